// dynamic_unfolding_4191888080915
// MI455X (gfx1250) — compile-verified
//
#include <hip/hip_runtime.h>
#include <hip/hip_bf16.h>

// ---------------------------------------------------------------------------
// out[y,x] = sm[y,x] * BoxFull8(local_count / BoxValid8(sm))[y,x]
//   sm = exp(mean_c x),  BoxValid8 = 8x8 'valid' box sum, BoxFull8 = overlap-add
// Pass 1: V_WMMA_F32_16X16X4_F32 channel reduction, 64 pixels/wave via
// 4 accumulator chains fed by b128 loads (pixel p = 4*n + chain).
// ---------------------------------------------------------------------------

typedef __attribute__((ext_vector_type(2))) float v2f;
typedef __attribute__((ext_vector_type(4))) float v4f;
typedef __attribute__((ext_vector_type(8))) float v8f;

#define AA 8
#define HH 256
#define WW 256
#define CC 64
#define KK 8
#define HP (HH - KK + 1)   // 249
#define WP (WW - KK + 1)   // 249
#define PIX (HH * WW)      // 65536 pixels per image
#define CH_STRIDE (HH * WW)

// ---- Pass 1: sm[a,p] = exp( (1/64) * sum_c x[a,c,p] ) via f32 WMMA ---------
// One wave computes 64 consecutive pixels (4 WMMA chains). 1024 waves/image.
__global__ void __launch_bounds__(256)
k_channel_mean_exp(const float* __restrict__ x, float* __restrict__ sm) {
    const int wave = (int)((blockIdx.x * blockDim.x + threadIdx.x) >> 5);
    const int lane = (int)(threadIdx.x & 31);
    const int img  = wave >> 10;              // 65536/64 = 1024 waves per image
    const int pix0 = (wave & 1023) << 6;      // base pixel of this wave's 64

    const int half = lane >> 4;               // 0: K=k0,k0+1   1: K=k0+2,k0+3
    const int n    = lane & 15;               // WMMA N index

    // Lane loads float4 covering pixels pix0 + 4n .. 4n+3 of one channel.
    const float* base = x + (size_t)img * CC * CH_STRIDE + pix0 + 4 * n;

    v2f a; a[0] = 1.0f; a[1] = 1.0f;          // ones A (16x4): D = colsum(B) + C
    v8f d0 = {}, d1 = {}, d2 = {}, d3 = {};

#pragma unroll
    for (int k0 = 0; k0 < CC; k0 += 4) {
        const float* p = base + (size_t)(k0 + 2 * half) * CH_STRIDE;
        v4f q0 = *(const v4f*)p;               // channel k0+2h, pixels 4n..4n+3
        v4f q1 = *(const v4f*)(p + CH_STRIDE); // channel k0+2h+1
        v2f b;
        b[0] = q0[0]; b[1] = q1[0];            // chain 0: pixels 4n+0
        d0 = __builtin_amdgcn_wmma_f32_16x16x4_f32(false, a, false, b, (short)0, d0, false, false);
        b[0] = q0[1]; b[1] = q1[1];            // chain 1: pixels 4n+1
        d1 = __builtin_amdgcn_wmma_f32_16x16x4_f32(false, a, false, b, (short)0, d1, false, false);
        b[0] = q0[2]; b[1] = q1[2];            // chain 2: pixels 4n+2
        d2 = __builtin_amdgcn_wmma_f32_16x16x4_f32(false, a, false, b, (short)0, d2, false, false);
        b[0] = q0[3]; b[1] = q1[3];            // chain 3: pixels 4n+3
        d3 = __builtin_amdgcn_wmma_f32_16x16x4_f32(false, a, false, b, (short)0, d3, false, false);
    }

    // D VGPR0 at lane l = channel sum for pixel (pix0 + 4*(l%16) + chain).
    const float scale = 1.0f / (float)CC;
    v4f o;
    o[0] = __expf(d0[0] * scale);
    o[1] = __expf(d1[0] * scale);
    o[2] = __expf(d2[0] * scale);
    o[3] = __expf(d3[0] * scale);
    if (half == 0)
        *(v4f*)(sm + (size_t)img * PIX + pix0 + 4 * n) = o;
}

// ---- Pass 2a: rowsum[a,y,xr] = sum_{j<8} sm[a,y,xr+j]  (8,256,249) ---------
__global__ void __launch_bounds__(256)
k_rowsum(const float* __restrict__ sm, float* __restrict__ rowsum) {
    int idx = blockIdx.x * blockDim.x + threadIdx.x;
    const int N = AA * HH * WP;
    if (idx >= N) return;
    int xr = idx % WP;
    int r  = idx / WP;
    int y  = r % HH;
    int a  = r / HH;
    const float* p = sm + ((size_t)a * HH + y) * WW + xr;
    float s = 0.f;
#pragma unroll
    for (int j = 0; j < KK; ++j) s += p[j];
    rowsum[idx] = s;
}

// ---- Pass 2b: t[a,yr,xr] = local_count / sum_{i<8} rowsum[a,yr+i,xr] -------
__global__ void __launch_bounds__(256)
k_patch_ratio(const float* __restrict__ rowsum, const float* __restrict__ lc,
              float* __restrict__ t) {
    int idx = blockIdx.x * blockDim.x + threadIdx.x;
    const int N = AA * HP * WP;
    if (idx >= N) return;
    int xr = idx % WP;
    int r  = idx / WP;
    int yr = r % HP;
    int a  = r / HP;
    float s = 0.f;
#pragma unroll
    for (int i = 0; i < KK; ++i)
        s += rowsum[((size_t)a * HH + yr + i) * WP + xr];
    t[idx] = lc[idx] / s;
}

// ---- Pass 3a: u[a,py,x] = sum_{px in [max(0,x-7),min(248,x)]} t[a,py,px] ---
__global__ void __launch_bounds__(256)
k_colspread(const float* __restrict__ t, float* __restrict__ u) {
    int idx = blockIdx.x * blockDim.x + threadIdx.x;
    const int N = AA * HP * WW;
    if (idx >= N) return;
    int x  = idx % WW;
    int r  = idx / WW;
    int py = r % HP;
    int a  = r / HP;
    int lo = x - (KK - 1); if (lo < 0) lo = 0;
    int hi = x; if (hi > WP - 1) hi = WP - 1;
    const float* row = t + ((size_t)a * HP + py) * WP;
    float s = 0.f;
    for (int px = lo; px <= hi; ++px) s += row[px];
    u[idx] = s;
}

// ---- Pass 3b: out[a,y,x] = sm * sum_{py in [max(0,y-7),min(248,y)]} u ------
__global__ void __launch_bounds__(256)
k_finalize(const float* __restrict__ u, const float* __restrict__ sm,
           float* __restrict__ out) {
    int idx = blockIdx.x * blockDim.x + threadIdx.x;
    const int N = AA * HH * WW;
    if (idx >= N) return;
    int x = idx & (WW - 1);
    int y = (idx >> 8) & (HH - 1);
    int a = idx >> 16;
    int lo = y - (KK - 1); if (lo < 0) lo = 0;
    int hi = y; if (hi > HP - 1) hi = HP - 1;
    float s = 0.f;
    for (int py = lo; py <= hi; ++py)
        s += u[((size_t)a * HP + py) * WW + x];
    out[idx] = sm[idx] * s;
}

extern "C" void kernel_launch(void* const* d_in, const int* in_sizes, int n_in,
                              void* d_out, int out_size, void* d_ws, size_t ws_size,
                              hipStream_t stream) {
    (void)in_sizes; (void)n_in; (void)out_size; (void)ws_size;

    const float* x  = (const float*)d_in[0];             // (8,64,256,256)
    const float* lc = (const float*)d_in[1];             // (8,1,249,249)
    float* out      = (float*)d_out;                     // (8,1,256,256)

    // Workspace layout (floats), ~6.1 MB total:
    float* ws      = (float*)d_ws;
    float* sm      = ws;                                  // 8*256*256 = 524288
    float* rowsum  = sm + (size_t)AA * PIX;               // 8*256*249 = 509952
    float* t       = rowsum + (size_t)AA * HH * WP;       // 8*249*249 = 496008
    float* u       = rowsum;                              // reuse (rowsum dead after 2b)

    // Pass 1: WMMA channel reduction + exp. 8192 waves -> 1024 blocks of 256.
    k_channel_mean_exp<<<(AA * PIX / 64) / 8, 256, 0, stream>>>(x, sm);

    // Pass 2: valid 8x8 box sum (separable) + divide.
    {
        int N = AA * HH * WP;
        k_rowsum<<<(N + 255) / 256, 256, 0, stream>>>(sm, rowsum);
    }
    {
        int N = AA * HP * WP;
        k_patch_ratio<<<(N + 255) / 256, 256, 0, stream>>>(rowsum, lc, t);
    }

    // Pass 3: overlap-add (full) 8x8 box sum (separable) and final multiply.
    {
        int N = AA * HP * WW;
        k_colspread<<<(N + 255) / 256, 256, 0, stream>>>(t, u);
    }
    {
        int N = AA * HH * WW;
        k_finalize<<<(N + 255) / 256, 256, 0, stream>>>(u, sm, out);
    }
}